// DNN_Model_64793876628183
// MI455X (gfx1250) — compile-verified
//
#include <hip/hip_runtime.h>
#include <math.h>

typedef __attribute__((ext_vector_type(2))) float v2f;
typedef __attribute__((ext_vector_type(8))) float v8f;

#define BATCH   128
#define C_SZ    64
#define NRIS    100
#define NELEM   (NRIS * NRIS)       // 10000
#define ODIM    264
#define INV_THRESH 1.0e15f          // 1 / THRESHOLD_W

#define LDA_PAD 18                  // 72B row stride: 8B-aligned v2f reads, no bank conflicts

// ---------------------------------------------------------------------------
// f32 WMMA GEMM:  out[M,N] = act(A[M,K] @ W[K,N] + bias[N])
// Block = 128 threads (4 waves) -> 16(M) x 64(N) tile, K stepped by 16.
// A tile (16x16, shared by all 4 waves) and B tile (16x64, stored transposed
// [col][k]) staged in LDS with coalesced, UNGUARDED global loads in the main
// loop: column tails are handled by clamping the load column (garbage results
// for col>=N are never stored), and the K tail is peeled into one guarded
// zero-filled stage. The WMMA loop is branch-free with EXEC all-ones; each
// fragment read is one ds_load_b64 (paired into ds_load_2addr_b64).
// ---------------------------------------------------------------------------
__global__ __launch_bounds__(128) void gemm_wmma_f32(
    const float* __restrict__ A, const float* __restrict__ W,
    const float* __restrict__ bias, float* __restrict__ out,
    int M, int N, int K, int relu)
{
    __shared__ float lA[16 * LDA_PAD];      // [m][k]
    __shared__ float lB[64 * LDA_PAD];      // [col][k]  (transposed)

    const int tid  = threadIdx.x;
    const int lane = tid & 31;
    const int wave = tid >> 5;
    const int half = lane >> 4;             // 0: K={0,1}, 1: K={2,3}
    const int l    = lane & 15;
    const int tileM = blockIdx.x * 16;
    const int nbase = blockIdx.y * 64;
    const int col   = nbase + wave * 16 + l;        // output column of this lane

    // staging roles
    const int ar = tid >> 4;                // 0..7  : A row pair (ar, ar+8)
    const int ac = tid & 15;                // 0..15 : A k-offset (coalesced)
    const int brow0 = tid >> 6;             // 0..1  : B k-row start
    const int bcol  = tid & 63;             // 0..63 : B column (coalesced)

    // clamped column for B staging loads: valid address, result unused if OOB
    const int gcL = min(nbase + bcol, N - 1);

    const float* __restrict__ Arow0 = A + (size_t)(tileM + ar) * K;
    const float* __restrict__ Arow8 = A + (size_t)(tileM + ar + 8) * K;

    v8f acc = {};

    const int kfull = K & ~15;
    int k = 0;
    for (; k < kfull; k += 16) {            // ---- main loop: branch-free ----
        __syncthreads();
        lA[ar * LDA_PAD + ac]       = Arow0[k + ac];
        lA[(ar + 8) * LDA_PAD + ac] = Arow8[k + ac];
        #pragma unroll
        for (int i = 0; i < 8; ++i) {
            const int rrow = brow0 + i * 2;           // 0..15
            lB[bcol * LDA_PAD + rrow] = W[(size_t)(k + rrow) * N + gcL];
        }
        __syncthreads();

        #pragma unroll
        for (int kk = 0; kk < 16; kk += 4) {
            const int k0 = kk + half * 2;
            v2f av = *(const v2f*)&lA[l * LDA_PAD + k0];
            v2f bv = *(const v2f*)&lB[(wave * 16 + l) * LDA_PAD + k0];
            acc = __builtin_amdgcn_wmma_f32_16x16x4_f32(
                false, av, false, bv, (short)0, acc, false, false);
        }
    }

    if (k < K) {                            // ---- K tail (runs <= once) ----
        __syncthreads();
        {
            const int gk = k + ac;
            const bool ok = (gk < K);
            lA[ar * LDA_PAD + ac]       = ok ? Arow0[gk] : 0.f;
            lA[(ar + 8) * LDA_PAD + ac] = ok ? Arow8[gk] : 0.f;
        }
        #pragma unroll
        for (int i = 0; i < 8; ++i) {
            const int rrow = brow0 + i * 2;
            const int gk = k + rrow;
            lB[bcol * LDA_PAD + rrow] = (gk < K) ? W[(size_t)gk * N + gcL] : 0.f;
        }
        __syncthreads();

        #pragma unroll
        for (int kk = 0; kk < 16; kk += 4) {
            const int k0 = kk + half * 2;
            v2f av = *(const v2f*)&lA[l * LDA_PAD + k0];
            v2f bv = *(const v2f*)&lB[(wave * 16 + l) * LDA_PAD + k0];
            acc = __builtin_amdgcn_wmma_f32_16x16x4_f32(
                false, av, false, bv, (short)0, acc, false, false);
        }
    }

    if (col < N) {
        const float bb = bias[col];
        #pragma unroll
        for (int v = 0; v < 8; ++v) {
            const int m = tileM + half * 8 + v;
            float val = acc[v] + bb;
            if (relu) val = fmaxf(val, 0.f);
            out[(size_t)m * N + col] = val;
        }
    }
}

// ---------------------------------------------------------------------------
// theta = (re + i*im) / |.|   -> separate re/im planes [B,100]
// ---------------------------------------------------------------------------
__global__ __launch_bounds__(128) void theta_normalize(
    const float* __restrict__ theta1,
    float* __restrict__ ta, float* __restrict__ tb)
{
    const int b = blockIdx.x;
    const int i = threadIdx.x;
    if (i < NRIS) {
        const float re = theta1[b * ODIM + i];
        const float im = theta1[b * ODIM + NRIS + i];
        const float inv = rsqrtf(re * re + im * im);
        ta[b * NRIS + i] = re * inv;
        tb[b * NRIS + i] = im * inv;
    }
}

// ---------------------------------------------------------------------------
// ccc[b,c] = Re(theta^H T theta)/THRESH
//          = sum_{n,m} Tr*(a_n a_m + b_n b_m) - Ti*(a_n b_m - b_n a_m)
// One block per (b,c); theta (800B) staged in LDS; T streamed as float4
// (global_load_b128) with global_prefetch_b8 lookahead. Memory-bound:
// 80 KB per block, 655 MB total -> ~28 us floor at 23.3 TB/s.
// ---------------------------------------------------------------------------
__global__ __launch_bounds__(256) void quad_form(
    const float* __restrict__ T_real, const float* __restrict__ T_imag,
    const float* __restrict__ ta, const float* __restrict__ tb,
    float* __restrict__ ccc)
{
    __shared__ float sa[NRIS];
    __shared__ float sb[NRIS];
    __shared__ float sred[8];

    const int bc  = blockIdx.x;          // b*64 + c
    const int b   = bc >> 6;
    const int tid = threadIdx.x;
    if (tid < NRIS) {
        sa[tid] = ta[b * NRIS + tid];
        sb[tid] = tb[b * NRIS + tid];
    }
    __syncthreads();

    const float4* __restrict__ Tr = (const float4*)(T_real + (size_t)bc * NELEM);
    const float4* __restrict__ Ti = (const float4*)(T_imag + (size_t)bc * NELEM);

    float acc = 0.f;
    for (int e4 = tid; e4 < NELEM / 4; e4 += 256) {
        __builtin_prefetch(Tr + e4 + 512, 0, 0);   // global_prefetch_b8
        __builtin_prefetch(Ti + e4 + 512, 0, 0);
        const int e = e4 * 4;
        const int n = e / NRIS;                    // 100 % 4 == 0: row-aligned
        const int m = e - n * NRIS;
        const float p = sa[n], q = sb[n];
        const float4 tr = Tr[e4];
        const float4 ti = Ti[e4];
        acc += tr.x * (p * sa[m + 0] + q * sb[m + 0]) - ti.x * (p * sb[m + 0] - q * sa[m + 0]);
        acc += tr.y * (p * sa[m + 1] + q * sb[m + 1]) - ti.y * (p * sb[m + 1] - q * sa[m + 1]);
        acc += tr.z * (p * sa[m + 2] + q * sb[m + 2]) - ti.z * (p * sb[m + 2] - q * sa[m + 2]);
        acc += tr.w * (p * sa[m + 3] + q * sb[m + 3]) - ti.w * (p * sb[m + 3] - q * sa[m + 3]);
    }

    // wave32 reduction, then cross-wave via LDS
    #pragma unroll
    for (int off = 16; off > 0; off >>= 1)
        acc += __shfl_xor(acc, off, 32);
    if ((tid & 31) == 0) sred[tid >> 5] = acc;
    __syncthreads();
    if (tid == 0) {
        float s = 0.f;
        #pragma unroll
        for (int w = 0; w < 8; ++w) s += sred[w];
        ccc[bc] = s * INV_THRESH;
    }
}

// ---------------------------------------------------------------------------
// scale[b] = rsqrt(max_c ccc[b,c]); write [re(theta_hat), im(theta_hat),
// theta1[:,200:264]] into d_out [128,264].
// ---------------------------------------------------------------------------
__global__ __launch_bounds__(128) void finalize_out(
    const float* __restrict__ ccc, const float* __restrict__ ta,
    const float* __restrict__ tb, const float* __restrict__ theta1,
    float* __restrict__ out)
{
    __shared__ float smax[2];
    __shared__ float sscale;
    const int b   = blockIdx.x;
    const int tid = threadIdx.x;

    float m = -INFINITY;
    if (tid < C_SZ) m = ccc[b * C_SZ + tid];
    #pragma unroll
    for (int off = 16; off > 0; off >>= 1)
        m = fmaxf(m, __shfl_xor(m, off, 32));
    if (tid < C_SZ && (tid & 31) == 0) smax[tid >> 5] = m;
    __syncthreads();
    if (tid == 0) sscale = rsqrtf(fmaxf(smax[0], smax[1]));
    __syncthreads();

    const float s = sscale;
    if (tid < NRIS) {
        out[b * ODIM + tid]        = ta[b * NRIS + tid] * s;
        out[b * ODIM + NRIS + tid] = tb[b * NRIS + tid] * s;
    }
    if (tid < ODIM - 2 * NRIS)     // 64 passthrough columns
        out[b * ODIM + 2 * NRIS + tid] = theta1[b * ODIM + 2 * NRIS + tid];
}

// ---------------------------------------------------------------------------
extern "C" void kernel_launch(void* const* d_in, const int* in_sizes, int n_in,
                              void* d_out, int out_size, void* d_ws, size_t ws_size,
                              hipStream_t stream)
{
    const float* sample1 = (const float*)d_in[0];
    // d_in[1] = sample2 : unused by the reference
    const float* T_real  = (const float*)d_in[2];
    const float* T_imag  = (const float*)d_in[3];
    const float* W1 = (const float*)d_in[4];   const float* b1 = (const float*)d_in[5];
    const float* W2 = (const float*)d_in[6];   const float* b2 = (const float*)d_in[7];
    const float* W3 = (const float*)d_in[8];   const float* b3 = (const float*)d_in[9];
    const float* W4 = (const float*)d_in[10];  const float* b4 = (const float*)d_in[11];
    const float* W5 = (const float*)d_in[12];  const float* b5 = (const float*)d_in[13];

    float* ws  = (float*)d_ws;
    float* h1  = ws;                          // 128*1024
    float* h2  = h1  + BATCH * 1024;          // 128*1024
    float* h3  = h2  + BATCH * 1024;          // 128*1024
    float* h4  = h3  + BATCH * 1024;          // 128*512
    float* th1 = h4  + BATCH * 512;           // 128*264
    float* ta  = th1 + BATCH * ODIM;          // 128*100
    float* tb  = ta  + BATCH * NRIS;          // 128*100
    float* ccc = tb  + BATCH * NRIS;          // 128*64

    const dim3 blk(128);
    // M=128 -> 8 tile-rows; grid.y covers N in 64-wide (4 waves) slabs
    gemm_wmma_f32<<<dim3(8, 16), blk, 0, stream>>>(sample1, W1, b1, h1, BATCH, 1024, 303, 1);
    gemm_wmma_f32<<<dim3(8, 16), blk, 0, stream>>>(h1, W2, b2, h2, BATCH, 1024, 1024, 1);
    gemm_wmma_f32<<<dim3(8, 16), blk, 0, stream>>>(h2, W3, b3, h3, BATCH, 1024, 1024, 1);
    gemm_wmma_f32<<<dim3(8,  8), blk, 0, stream>>>(h3, W4, b4, h4, BATCH, 512, 1024, 1);
    gemm_wmma_f32<<<dim3(8,  5), blk, 0, stream>>>(h4, W5, b5, th1, BATCH, ODIM, 512, 0);

    theta_normalize<<<dim3(BATCH), dim3(128), 0, stream>>>(th1, ta, tb);
    quad_form<<<dim3(BATCH * C_SZ), dim3(256), 0, stream>>>(T_real, T_imag, ta, tb, ccc);
    finalize_out<<<dim3(BATCH), dim3(128), 0, stream>>>(ccc, ta, tb, th1, (float*)d_out);
}